// GCNModel_43714177138844
// MI455X (gfx1250) — compile-verified
//
#include <hip/hip_runtime.h>
#include <hip/hip_bf16.h>

typedef __attribute__((ext_vector_type(2))) float v2f;
typedef __attribute__((ext_vector_type(8))) float v8f;

#define HID 16

// ---------------------------------------------------------------------------
// Init: deg = 1.0 (self-loop), zero the two scatter accumulators.
// ---------------------------------------------------------------------------
__global__ __launch_bounds__(256) void k_init(float* __restrict__ deg,
                                              float* __restrict__ B,
                                              float* __restrict__ C,
                                              int n_nodes) {
  int i = blockIdx.x * blockDim.x + threadIdx.x;
  if (i < n_nodes) deg[i] = 1.0f;
  if (i < n_nodes * HID) { B[i] = 0.0f; C[i] = 0.0f; }
}

// ---------------------------------------------------------------------------
// In-degree via fp32 atomics (accumulator lives in L2), then rsqrt in place.
// ---------------------------------------------------------------------------
__global__ __launch_bounds__(256) void k_degree(const int* __restrict__ dst,
                                                float* __restrict__ deg,
                                                int n_edges) {
  int e = blockIdx.x * blockDim.x + threadIdx.x;
  if (e < n_edges) atomicAdd(&deg[dst[e]], 1.0f);
}

__global__ __launch_bounds__(256) void k_dinv(float* __restrict__ deg, int n_nodes) {
  int i = blockIdx.x * blockDim.x + threadIdx.x;
  if (i < n_nodes) deg[i] = rsqrtf(deg[i]);   // deg >= 1 always (self-loop)
}

// ---------------------------------------------------------------------------
// Tall-skinny GEMM:  out[M x 16] = A[M x K] @ W[K x 16]   (fp32 WMMA 16x16x4)
// One wave32 owns a 16-row tile of nodes.
//   A frag : lane half h = lane>>4 supplies K = kk+2h, kk+2h+1 for row lane&15
//   B frag : lane half supplies same K pair for column lane&15
//   C/D    : VGPR j -> row j (+8 on upper half), col = lane&15
// ---------------------------------------------------------------------------
__global__ __launch_bounds__(256) void k_gemm16(const float* __restrict__ Amat,
                                                const float* __restrict__ W,
                                                float* __restrict__ out,
                                                int n_nodes, int K) {
  int wave = (int)((blockIdx.x * (unsigned)blockDim.x + threadIdx.x) >> 5);
  int lane = threadIdx.x & 31;
  int ntiles = (n_nodes + 15) >> 4;
  if (wave >= ntiles) return;            // wave-uniform: EXEC stays all-1s

  int half = lane >> 4;                  // 0: lanes 0-15, 1: lanes 16-31
  int l16  = lane & 15;
  int row  = wave * 16 + l16;
  int rowc = row < n_nodes ? row : n_nodes - 1;   // clamp keeps EXEC full

  v8f c = {};
  for (int kk = 0; kk < K; kk += 4) {
    int ks = kk + half * 2;
    v2f a, b;
    a[0] = Amat[(size_t)rowc * K + ks];
    a[1] = Amat[(size_t)rowc * K + ks + 1];
    b[0] = W[(ks + 0) * HID + l16];
    b[1] = W[(ks + 1) * HID + l16];
    c = __builtin_amdgcn_wmma_f32_16x16x4_f32(false, a, false, b,
                                              (short)0, c, false, false);
  }

  int rbase = wave * 16 + half * 8;
  int col   = l16;
#pragma unroll
  for (int j = 0; j < 8; ++j) {
    int r = rbase + j;
    if (r < n_nodes) out[(size_t)r * HID + col] = c[j];
  }
}

// ---------------------------------------------------------------------------
// Edge scatter: out[dst] += hlin[src] * dinv[src]*dinv[dst]
// 16 consecutive lanes = 16 features of one edge -> coalesced 64B gathers and
// coalesced global_atomic_add_f32 bursts (resolve in L2; 6.4MB accumulator).
// ---------------------------------------------------------------------------
__global__ __launch_bounds__(256) void k_scatter(const float* __restrict__ hlin,
                                                 const int* __restrict__ src,
                                                 const int* __restrict__ dst,
                                                 const float* __restrict__ dinv,
                                                 float* __restrict__ out,
                                                 int n_edges) {
  int gid = blockIdx.x * blockDim.x + threadIdx.x;
  int e = gid >> 4;
  int f = gid & 15;
  if (e >= n_edges) return;
  int s = src[e], d = dst[e];
  float norm = dinv[s] * dinv[d];
  atomicAdd(&out[(size_t)d * HID + f], hlin[(size_t)s * HID + f] * norm);
}

// ---------------------------------------------------------------------------
// Self-loop term (dinv[i]^2 == 1/deg[i]) + bias (+ optional ReLU), in place.
// ---------------------------------------------------------------------------
__global__ __launch_bounds__(256) void k_finish(const float* __restrict__ hlin,
                                                const float* __restrict__ dinv,
                                                const float* __restrict__ bias,
                                                float* __restrict__ out,
                                                int n_nodes, int do_relu) {
  int gid = blockIdx.x * blockDim.x + threadIdx.x;
  int i = gid >> 4;
  int f = gid & 15;
  if (i >= n_nodes) return;
  float di = dinv[i];
  float v = out[(size_t)i * HID + f] + hlin[(size_t)i * HID + f] * di * di + bias[f];
  if (do_relu) v = fmaxf(v, 0.0f);
  out[(size_t)i * HID + f] = v;
}

// ---------------------------------------------------------------------------
// Edge predictor: pred[e] = [h2[src], h2[dst]] . Wf + bf  (K=32 GEMV, gather-
// bound -> VALU; float4 loads, Wf/bf are uniform -> scalar-cached).
// ---------------------------------------------------------------------------
__global__ __launch_bounds__(256) void k_predict(const float* __restrict__ h2,
                                                 const int* __restrict__ src,
                                                 const int* __restrict__ dst,
                                                 const float* __restrict__ Wf,
                                                 const float* __restrict__ bf,
                                                 float* __restrict__ pred,
                                                 int n_edges) {
  int e = blockIdx.x * blockDim.x + threadIdx.x;
  if (e >= n_edges) return;
  int s = src[e], d = dst[e];
  const float4* hs = (const float4*)(h2 + (size_t)s * HID);
  const float4* hd = (const float4*)(h2 + (size_t)d * HID);
  float acc = bf[0];
#pragma unroll
  for (int q = 0; q < 4; ++q) {
    float4 v = hs[q];
    acc += v.x * Wf[q * 4 + 0] + v.y * Wf[q * 4 + 1] +
           v.z * Wf[q * 4 + 2] + v.w * Wf[q * 4 + 3];
  }
#pragma unroll
  for (int q = 0; q < 4; ++q) {
    float4 v = hd[q];
    acc += v.x * Wf[16 + q * 4 + 0] + v.y * Wf[16 + q * 4 + 1] +
           v.z * Wf[16 + q * 4 + 2] + v.w * Wf[16 + q * 4 + 3];
  }
  pred[e] = acc;
}

// ---------------------------------------------------------------------------
extern "C" void kernel_launch(void* const* d_in, const int* in_sizes, int n_in,
                              void* d_out, int out_size, void* d_ws, size_t ws_size,
                              hipStream_t stream) {
  const float* x  = (const float*)d_in[0];
  const int*   ei = (const int*)d_in[1];
  const float* W1 = (const float*)d_in[2];
  const float* b1 = (const float*)d_in[3];
  const float* W2 = (const float*)d_in[4];
  const float* b2 = (const float*)d_in[5];
  const float* Wf = (const float*)d_in[6];
  const float* bf = (const float*)d_in[7];
  float* pred = (float*)d_out;

  const int hid     = in_sizes[3];            // 16
  const int f_in    = in_sizes[2] / hid;      // 128
  const int n_nodes = in_sizes[0] / f_in;     // 100000
  const int n_edges = in_sizes[1] / 2;        // 3200000
  const int* src = ei;
  const int* dst = ei + n_edges;

  // Workspace carve-up (256B aligned): dinv | A (lin out) | B (h1) | C (h2)
  char* ws = (char*)d_ws;
  size_t off = 0;
  float* dinv = (float*)(ws + off);
  off += (((size_t)n_nodes * 4) + 255) & ~(size_t)255;
  float* A = (float*)(ws + off);
  off += (((size_t)n_nodes * HID * 4) + 255) & ~(size_t)255;
  float* B = (float*)(ws + off);
  off += (((size_t)n_nodes * HID * 4) + 255) & ~(size_t)255;
  float* C = (float*)(ws + off);

  const int nf = n_nodes * HID;               // 1.6M
  const int ef = n_edges * HID;               // 51.2M (< 2^31)
  dim3 blk(256);
  int ntiles = (n_nodes + 15) / 16;
  int gemm_blocks = (ntiles + 7) / 8;         // 8 waves per 256-thread block

  // 0) init: deg=1 (self-loop), zero accumulators
  k_init<<<(nf + 255) / 256, blk, 0, stream>>>(dinv, B, C, n_nodes);
  // 1) in-degree + rsqrt
  k_degree<<<(n_edges + 255) / 256, blk, 0, stream>>>(dst, dinv, n_edges);
  k_dinv<<<(n_nodes + 255) / 256, blk, 0, stream>>>(dinv, n_nodes);
  // 2) layer 1: A = x @ W1   (WMMA f32, K=128)
  k_gemm16<<<gemm_blocks, blk, 0, stream>>>(x, W1, A, n_nodes, f_in);
  // 3) scatter + self-loop + b1 + ReLU  -> B
  k_scatter<<<(ef + 255) / 256, blk, 0, stream>>>(A, src, dst, dinv, B, n_edges);
  k_finish<<<(nf + 255) / 256, blk, 0, stream>>>(A, dinv, b1, B, n_nodes, 1);
  // 4) layer 2: A = B @ W2   (WMMA f32, K=16)
  k_gemm16<<<gemm_blocks, blk, 0, stream>>>(B, W2, A, n_nodes, hid);
  // 5) scatter + self-loop + b2 (no ReLU) -> C
  k_scatter<<<(ef + 255) / 256, blk, 0, stream>>>(A, src, dst, dinv, C, n_edges);
  k_finish<<<(nf + 255) / 256, blk, 0, stream>>>(A, dinv, b2, C, n_nodes, 0);
  // 6) edge predictor
  k_predict<<<(n_edges + 255) / 256, blk, 0, stream>>>(C, src, dst, Wf, bf, pred, n_edges);
  (void)ws_size; (void)n_in; (void)out_size;
}